// SpLearner_36696200577528
// MI455X (gfx1250) — compile-verified
//
#include <hip/hip_runtime.h>
#include <hip/hip_bf16.h>
#include <math.h>

// ---------------------------------------------------------------------------
// SpLearner edge scorer for MI455X (gfx1250, wave32, WMMA)
//   phase 1: per-edge MLP via V_WMMA_F32_16X16X4_F32 (one wave = one node's
//            16-edge tile, K padded to 132, N=64 -> 132 WMMAs / wave)
//   phase 2: deterministic ||z|| reduction
//   phase 3: per-node double softmax + gumbel + top-8 threshold mask
// ---------------------------------------------------------------------------

#define N_NODES   100000
#define DEG       16
#define DFEAT     64
#define HID       64
#define IN_DIM    129      // 2*DFEAT + 1
#define KCH       33       // ceil(132/4) K-chunks of 4
#define NT        4        // 64 / 16 N tiles
#define FRAG_B_FLOATS (KCH * NT * 32 * 2)   // 8448 floats, swizzled B fragments

typedef __attribute__((ext_vector_type(2))) float v2f;
typedef __attribute__((ext_vector_type(8))) float v8f;

#if __has_builtin(__builtin_amdgcn_wmma_f32_16x16x4_f32)
#define HAVE_WMMA_F32X4 1
#endif

static __device__ __forceinline__ v8f wmma_f32x4(v2f a, v2f b, v8f c) {
#ifdef HAVE_WMMA_F32X4
    // 8 args: (neg_a, A, neg_b, B, c_mod, C, reuse_a, reuse_b)
    return __builtin_amdgcn_wmma_f32_16x16x4_f32(
        false, a, false, b, (short)0, c, false, false);
#else
    asm volatile("v_wmma_f32_16x16x4_f32 %0, %1, %2, %0"
                 : "+v"(c) : "v"(a), "v"(b));
    return c;
#endif
}

// ---------------------------------------------------------------------------
// Pre-swizzle w0 (HID x IN_DIM, row-major) into WMMA B fragments:
//   fragB[((c*NT + t)*32 + lane)*2 + d] = w0T_pad[4c + 2*half + d][16t + lm]
// where half = lane>>4, lm = lane&15, w0T_pad[k][n] = (k<129 ? w0[n][k] : 0)
// ---------------------------------------------------------------------------
__global__ __launch_bounds__(256) void splearner_prep_fragB(
    const float* __restrict__ w0, float* __restrict__ fragB)
{
    int i = blockIdx.x * 256 + threadIdx.x;
    if (i >= FRAG_B_FLOATS) return;
    int d    = i & 1;
    int lane = (i >> 1) & 31;
    int t    = (i >> 6) & 3;
    int c    = i >> 8;
    int half = lane >> 4, lm = lane & 15;
    int k = c * 4 + 2 * half + d;
    int n = t * 16 + lm;
    fragB[i] = (k < IN_DIM) ? w0[n * IN_DIM + k] : 0.0f;
}

// ---------------------------------------------------------------------------
// Phase 1: one wave per node. A-tile = 16 edges x 132 (row feats | col feats
// | value | pad). 33 K-chunks x 4 N-tiles of v_wmma_f32_16x16x4_f32, then
// bias+relu+second layer in-register, shfl_xor butterfly to form z[16].
// ---------------------------------------------------------------------------
__global__ __launch_bounds__(256) void splearner_mlp_wmma(
    const float* __restrict__ features,   // [N, 64]
    const float* __restrict__ values,     // [E]
    const float* __restrict__ p_b0,       // [64]
    const float* __restrict__ p_w1,       // [64]
    const float* __restrict__ p_b1,       // [1]
    const int*   __restrict__ cols,       // [E]
    const float* __restrict__ fragB,      // [FRAG_B_FLOATS]
    float*       __restrict__ z)          // [E]
{
    __shared__ __align__(16) float sB[FRAG_B_FLOATS];
    for (int i = threadIdx.x; i < FRAG_B_FLOATS; i += 256)
        sB[i] = fragB[i];
    __syncthreads();

    const int wave = threadIdx.x >> 5;
    const int lane = threadIdx.x & 31;
    const int half = lane >> 4;
    const int lm   = lane & 15;
    const int node = blockIdx.x * 8 + wave;
    if (node >= N_NODES) return;
    const int ebase = node * DEG;

    const int   colid = cols[ebase + lm];            // lane's M-row neighbor
    const float val   = values[ebase + lm];
    const float* Frow = features + (size_t)node  * DFEAT;
    const float* Fcol = features + (size_t)colid * DFEAT;

    v8f acc0 = {}, acc1 = {}, acc2 = {}, acc3 = {};

    // K-chunks 0..15: row-feature columns 0..63 (wave-uniform broadcast)
#pragma unroll
    for (int c = 0; c < 16; ++c) {
        const int k = c * 4 + half * 2;
        v2f a = *(const v2f*)(Frow + k);
        const float* bp = sB + c * 256 + lane * 2;
        acc0 = wmma_f32x4(a, *(const v2f*)(bp +   0), acc0);
        acc1 = wmma_f32x4(a, *(const v2f*)(bp +  64), acc1);
        acc2 = wmma_f32x4(a, *(const v2f*)(bp + 128), acc2);
        acc3 = wmma_f32x4(a, *(const v2f*)(bp + 192), acc3);
    }
    // K-chunks 16..31: col-feature columns 64..127 (per-M gather)
#pragma unroll
    for (int c = 16; c < 32; ++c) {
        const int k = (c - 16) * 4 + half * 2;
        v2f a = *(const v2f*)(Fcol + k);
        const float* bp = sB + c * 256 + lane * 2;
        acc0 = wmma_f32x4(a, *(const v2f*)(bp +   0), acc0);
        acc1 = wmma_f32x4(a, *(const v2f*)(bp +  64), acc1);
        acc2 = wmma_f32x4(a, *(const v2f*)(bp + 128), acc2);
        acc3 = wmma_f32x4(a, *(const v2f*)(bp + 192), acc3);
    }
    // K-chunk 32: column 128 = edge value, columns 129..131 = zero pad
    {
        v2f a; a.x = half ? 0.0f : val; a.y = 0.0f;
        const float* bp = sB + 32 * 256 + lane * 2;
        acc0 = wmma_f32x4(a, *(const v2f*)(bp +   0), acc0);
        acc1 = wmma_f32x4(a, *(const v2f*)(bp +  64), acc1);
        acc2 = wmma_f32x4(a, *(const v2f*)(bp + 128), acc2);
        acc3 = wmma_f32x4(a, *(const v2f*)(bp + 192), acc3);
    }

    // C/D layout: VGPR r holds (M = r + 8*half, N = 16*t + lm).
    // bias + relu + layer-2 partial dot per lane, then sum over the 16 lanes
    // of each half-wave (N dimension) with a shfl_xor butterfly.
    float part[8];
#pragma unroll
    for (int r = 0; r < 8; ++r) part[r] = 0.0f;
    {
        union { v8f v; float f[8]; } u;
        const v8f accs[4] = { acc0, acc1, acc2, acc3 };
#pragma unroll
        for (int t = 0; t < 4; ++t) {
            const float bb = p_b0[t * 16 + lm];
            const float ww = p_w1[t * 16 + lm];
            u.v = accs[t];
#pragma unroll
            for (int r = 0; r < 8; ++r) {
                float h = fmaxf(u.f[r] + bb, 0.0f);
                part[r] = fmaf(h, ww, part[r]);
            }
        }
    }
#pragma unroll
    for (int r = 0; r < 8; ++r) {
        float p = part[r];
        p += __shfl_xor(p, 1);
        p += __shfl_xor(p, 2);
        p += __shfl_xor(p, 4);
        p += __shfl_xor(p, 8);
        part[r] = p;
    }
    if (lm == 0) {
        const float b1v = p_b1[0];
#pragma unroll
        for (int r = 0; r < 8; ++r)
            z[ebase + half * 8 + r] = part[r] + b1v;
    }
}

// ---------------------------------------------------------------------------
// Phase 2: deterministic ||z||  (fixed-order chunk partials + 1-block tree)
// ---------------------------------------------------------------------------
#define SUMSQ_BLOCKS 200

__global__ __launch_bounds__(256) void splearner_sumsq_partial(
    const float* __restrict__ z, float* __restrict__ partial, int E, int chunk)
{
    __shared__ float s[256];
    const int base = blockIdx.x * chunk;
    float acc = 0.0f;
    for (int i = threadIdx.x; i < chunk && base + i < E; i += 256) {
        const float v = z[base + i];
        acc = fmaf(v, v, acc);
    }
    s[threadIdx.x] = acc;
    __syncthreads();
    for (int o = 128; o > 0; o >>= 1) {
        if (threadIdx.x < o) s[threadIdx.x] += s[threadIdx.x + o];
        __syncthreads();
    }
    if (threadIdx.x == 0) partial[blockIdx.x] = s[0];
}

__global__ __launch_bounds__(256) void splearner_finalize_norm(
    const float* __restrict__ partial, float* __restrict__ nrm)
{
    __shared__ float s[256];
    s[threadIdx.x] = (threadIdx.x < SUMSQ_BLOCKS) ? partial[threadIdx.x] : 0.0f;
    __syncthreads();
    for (int o = 128; o > 0; o >>= 1) {
        if (threadIdx.x < o) s[threadIdx.x] += s[threadIdx.x + o];
        __syncthreads();
    }
    if (threadIdx.x == 0) nrm[0] = sqrtf(s[0]);
}

// ---------------------------------------------------------------------------
// Phase 3: per-node (1 thread = 16 edges): normalize, softmax -> log + gumbel
// -> softmax, 8th-largest threshold (stable ties), binary-step mask.
// ---------------------------------------------------------------------------
__global__ __launch_bounds__(256) void splearner_mask(
    const float* __restrict__ z, const float* __restrict__ gumbel,
    const float* __restrict__ nrm, const int* __restrict__ temperature,
    float* __restrict__ out)
{
    const int node = blockIdx.x * 256 + threadIdx.x;
    if (node >= N_NODES) return;
    const float rT = 1.0f / (float)temperature[0];
    const float rn = 1.0f / fmaxf(nrm[0], 1e-12f);
    const float* zp = z + node * DEG;
    const float* gp = gumbel + node * DEG;

    float v[DEG];
    float m1 = -3.402823466e38f;
#pragma unroll
    for (int j = 0; j < DEG; ++j) { v[j] = zp[j] * rn; m1 = fmaxf(m1, v[j]); }
    float s1 = 0.0f;
#pragma unroll
    for (int j = 0; j < DEG; ++j) s1 += expf(v[j] - m1);
    const float lse = m1 + logf(s1);               // log(pi) = zn - lse

    float m2 = -3.402823466e38f;
#pragma unroll
    for (int j = 0; j < DEG; ++j) {
        v[j] = (v[j] - lse + gp[j]) * rT;          // gumbel logits
        m2 = fmaxf(m2, v[j]);
    }
    float s2 = 0.0f;
#pragma unroll
    for (int j = 0; j < DEG; ++j) { v[j] = expf(v[j] - m2); s2 += v[j]; }
#pragma unroll
    for (int j = 0; j < DEG; ++j) v[j] = v[j] / s2;   // y

    // 8th largest with stable tie order (first index among equals wins)
    float t[DEG];
#pragma unroll
    for (int j = 0; j < DEG; ++j) t[j] = v[j];
    float thre = 0.0f;
#pragma unroll
    for (int p = 0; p < 8; ++p) {
        int bi = 0; float bv = t[0];
#pragma unroll
        for (int j = 1; j < DEG; ++j)
            if (t[j] > bv) { bv = t[j]; bi = j; }
        thre = bv;
        t[bi] = -3.402823466e38f;
    }
#pragma unroll
    for (int j = 0; j < DEG; ++j)
        out[node * DEG + j] = ((v[j] - thre + 1e-12f) > 0.0f) ? v[j] : 0.0f;
}

// ---------------------------------------------------------------------------
extern "C" void kernel_launch(void* const* d_in, const int* in_sizes, int n_in,
                              void* d_out, int out_size, void* d_ws, size_t ws_size,
                              hipStream_t stream) {
    const float* features = (const float*)d_in[0];   // [N,64]
    const float* values   = (const float*)d_in[1];   // [E]
    const float* w0       = (const float*)d_in[2];   // [64,129]
    const float* b0       = (const float*)d_in[3];   // [64]
    const float* w1       = (const float*)d_in[4];   // [1,64]
    const float* b1       = (const float*)d_in[5];   // [1]
    const float* gumbel   = (const float*)d_in[6];   // [E]
    const int*   indices  = (const int*)d_in[7];     // [2,E]
    const int*   temp     = (const int*)d_in[9];     // scalar

    const int E = in_sizes[1];                       // 1,600,000
    const int* cols = indices + E;

    // workspace layout
    char* ws = (char*)d_ws;
    float* fragB   = (float*)ws;                                   // 8448 f
    float* z       = (float*)(ws + ((FRAG_B_FLOATS * 4 + 255) & ~255));
    float* partial = (float*)((char*)z + (size_t)E * 4);           // 200 f
    float* nrm     = partial + 256;                                // 1 f

    splearner_prep_fragB<<<(FRAG_B_FLOATS + 255) / 256, 256, 0, stream>>>(w0, fragB);

    splearner_mlp_wmma<<<N_NODES / 8, 256, 0, stream>>>(
        features, values, b0, w1, b1, cols, fragB, z);

    const int chunk = (E + SUMSQ_BLOCKS - 1) / SUMSQ_BLOCKS;
    splearner_sumsq_partial<<<SUMSQ_BLOCKS, 256, 0, stream>>>(z, partial, E, chunk);
    splearner_finalize_norm<<<1, 256, 0, stream>>>(partial, nrm);

    splearner_mask<<<(N_NODES + 255) / 256, 256, 0, stream>>>(
        z, gumbel, nrm, temp, (float*)d_out);
}